// RelationalCoherenceLossV2_64192581206702
// MI455X (gfx1250) — compile-verified
//
#include <hip/hip_runtime.h>
#include <hip/hip_bf16.h>
#include <math.h>
#include <stdint.h>

#define TPB 256
#define IGNORE_INDEX (-100)
#define LAMBDA_PRESENCE 0.15f
#define LAMBDA_COHERENCE 0.1f
#define MAX_V 32000   // V for this problem; row staged fully in LDS (125 KB of 320 KB/WGP)

// ---------------------------------------------------------------------------
// Kernel 1: one block per (b,s) token row. Stage the 32000-float row into LDS
// via CDNA5 async-DMA (global_load_async_to_lds_b128, ASYNCcnt), then compute
// logsumexp with a max pass + sum-exp pass over LDS, and emit 5 per-token
// partial values into the workspace.
//
// All strided loops are split into a uniform counted part (no divergence, no
// per-iteration exec-mask churn) plus a single guarded tail iteration.
// ---------------------------------------------------------------------------
__global__ __launch_bounds__(TPB) void rcl_row_kernel(
    const float* __restrict__ logits,
    const int* __restrict__ labels,
    const unsigned char* __restrict__ bond_mask,
    const unsigned char* __restrict__ rupture_mask,
    const float* __restrict__ presence,
    float* __restrict__ pc,   // ce * valid
    float* __restrict__ pw,   // ce * weight * valid
    float* __restrict__ pv,   // valid (0/1)
    float* __restrict__ pb,   // is_bond ? ps : 0
    float* __restrict__ pr,   // is_rup  ? ps : 0
    int V, int S)
{
  __shared__ __align__(16) float row[MAX_V];
  __shared__ float red[16];

  const int token = blockIdx.x;
  const unsigned tid = threadIdx.x;
  const float* base = logits + (size_t)token * (size_t)V;
  const int nch  = V >> 2;          // number of float4 chunks (8000)
  const int full = nch / TPB;       // uniform iterations (31)
  const int rem  = nch - full * TPB;// tail chunks (64)

  // ---- async DMA: global row -> LDS (no VGPR data path, tracked by ASYNCcnt)
  const unsigned lds_base = (unsigned)(uintptr_t)(&row[0]);  // low 32 bits = LDS byte addr
  {
    unsigned lds_addr = lds_base + tid * 16u;
    unsigned goff     = tid * 16u;           // GVS mode: saddr(base) + vaddr(offset)
#pragma unroll 4
    for (int k = 0; k < full; ++k) {
      asm volatile("global_load_async_to_lds_b128 %0, %1, %2"
                   :: "v"(lds_addr), "v"(goff), "s"(base)
                   : "memory");
      lds_addr += TPB * 16u;
      goff     += TPB * 16u;
    }
    if ((int)tid < rem) {
      asm volatile("global_load_async_to_lds_b128 %0, %1, %2"
                   :: "v"(lds_addr), "v"(goff), "s"(base)
                   : "memory");
    }
  }
  asm volatile("s_wait_asynccnt 0" ::: "memory");
  __syncthreads();

  // ---- pass 1: row max (float4 LDS reads)
  const float4* r4 = (const float4*)row;
  float m = -INFINITY;
#pragma unroll 4
  for (int k = 0; k < full; ++k) {
    float4 x = r4[tid + k * TPB];
    m = fmaxf(m, fmaxf(fmaxf(x.x, x.y), fmaxf(x.z, x.w)));
  }
  if ((int)tid < rem) {
    float4 x = r4[tid + full * TPB];
    m = fmaxf(m, fmaxf(fmaxf(x.x, x.y), fmaxf(x.z, x.w)));
  }
  for (int off = 16; off > 0; off >>= 1)
    m = fmaxf(m, __shfl_xor(m, off, 32));
  const unsigned wid = tid >> 5, lane = tid & 31;
  if (lane == 0) red[wid] = m;
  __syncthreads();
  float rowmax = red[0];
#pragma unroll
  for (int w = 1; w < 8; ++w) rowmax = fmaxf(rowmax, red[w]);

  // ---- pass 2: sum of exp(x - max)  (v_exp_f32 is TRANS, co-issues with VALU)
  float s = 0.f;
#pragma unroll 4
  for (int k = 0; k < full; ++k) {
    float4 x = r4[tid + k * TPB];
    s += __expf(x.x - rowmax) + __expf(x.y - rowmax) +
         __expf(x.z - rowmax) + __expf(x.w - rowmax);
  }
  if ((int)tid < rem) {
    float4 x = r4[tid + full * TPB];
    s += __expf(x.x - rowmax) + __expf(x.y - rowmax) +
         __expf(x.z - rowmax) + __expf(x.w - rowmax);
  }
  for (int off = 16; off > 0; off >>= 1)
    s += __shfl_xor(s, off, 32);
  if (lane == 0) red[8 + wid] = s;
  __syncthreads();

  if (tid == 0) {
    float rowsum = 0.f;
#pragma unroll
    for (int w = 0; w < 8; ++w) rowsum += red[8 + w];
    const float lse = rowmax + logf(rowsum);

    const int lab = labels[token];
    const bool valid = (lab != IGNORE_INDEX);
    const int safe = valid ? lab : 0;
    const float tgt = row[safe];          // gather is free: row already in LDS
    const float ce = lse - tgt;

    const int b = token / S;
    const float ps = presence[b];
    const bool bnd = bond_mask[safe] != 0;
    const bool rp  = rupture_mask[safe] != 0;
    const bool active = (ps > 0.f) && valid;
    const bool is_bond = bnd && active;
    const bool is_rup  = rp && !bnd && active;

    const float w = 1.f + (is_bond ? LAMBDA_PRESENCE * ps : 0.f)
                        + (is_rup  ? LAMBDA_COHERENCE * ps * 2.0f : 0.f);
    const float vf = valid ? 1.f : 0.f;

    pc[token] = ce * vf;
    pw[token] = ce * w * vf;
    pv[token] = vf;
    pb[token] = is_bond ? ps : 0.f;
    pr[token] = is_rup  ? ps : 0.f;
  }
}

// ---------------------------------------------------------------------------
// Kernel 2: deterministic reduction of the 5 per-token partial arrays and
// final scalar math. Single block, fixed reduction tree -> bit-reproducible.
// ntok is assumed to be a multiple of 4*TPB (4096 here); float4 loads.
// ---------------------------------------------------------------------------
__global__ __launch_bounds__(TPB) void rcl_final_kernel(
    const float* __restrict__ pc, const float* __restrict__ pw,
    const float* __restrict__ pv, const float* __restrict__ pb,
    const float* __restrict__ pr, float* __restrict__ out,
    int ntok, float inv_bsz)
{
  float sc = 0.f, sw = 0.f, sv = 0.f, sb = 0.f, sr = 0.f;
  const int n4 = ntok >> 2;
  const float4* c4 = (const float4*)pc;
  const float4* w4 = (const float4*)pw;
  const float4* v4 = (const float4*)pv;
  const float4* b4 = (const float4*)pb;
  const float4* r4 = (const float4*)pr;
  for (int i = (int)threadIdx.x; i < n4; i += TPB) {
    float4 x;
    x = c4[i]; sc += (x.x + x.y) + (x.z + x.w);
    x = w4[i]; sw += (x.x + x.y) + (x.z + x.w);
    x = v4[i]; sv += (x.x + x.y) + (x.z + x.w);
    x = b4[i]; sb += (x.x + x.y) + (x.z + x.w);
    x = r4[i]; sr += (x.x + x.y) + (x.z + x.w);
  }
  for (int off = 16; off > 0; off >>= 1) {
    sc += __shfl_xor(sc, off, 32);
    sw += __shfl_xor(sw, off, 32);
    sv += __shfl_xor(sv, off, 32);
    sb += __shfl_xor(sb, off, 32);
    sr += __shfl_xor(sr, off, 32);
  }
  __shared__ float red[8][5];
  const unsigned wid = threadIdx.x >> 5, lane = threadIdx.x & 31;
  if (lane == 0) {
    red[wid][0] = sc; red[wid][1] = sw; red[wid][2] = sv;
    red[wid][3] = sb; red[wid][4] = sr;
  }
  __syncthreads();
  if (threadIdx.x == 0) {
    float tc = 0.f, tw = 0.f, tv = 0.f, tb = 0.f, tr = 0.f;
#pragma unroll
    for (int w = 0; w < 8; ++w) {
      tc += red[w][0]; tw += red[w][1]; tv += red[w][2];
      tb += red[w][3]; tr += red[w][4];
    }
    const float denom = fmaxf(tv, 1.0f);
    const float lm  = (tv > 0.f) ? tc / denom : 0.f;
    const float tot = (tv > 0.f) ? tw / denom : 0.f;
    out[0] = tot;            // total_loss
    out[1] = lm;             // lm_loss
    out[2] = tb * inv_bsz;   // presence_loss
    out[3] = tot - lm;       // coherence_loss
    out[4] = tr * inv_bsz;   // continuity_loss
  }
}

extern "C" void kernel_launch(void* const* d_in, const int* in_sizes, int n_in,
                              void* d_out, int out_size, void* d_ws, size_t ws_size,
                              hipStream_t stream) {
  const float*         logits = (const float*)d_in[0];
  const int*           labels = (const int*)d_in[1];
  const unsigned char* bond   = (const unsigned char*)d_in[2];  // jnp bool -> 1 byte
  const unsigned char* rup    = (const unsigned char*)d_in[3];
  const float*         ps     = (const float*)d_in[4];
  float* out = (float*)d_out;

  const int ntok = in_sizes[1];       // B*S = 4096
  const int V    = in_sizes[2];       // 32000
  const int B    = in_sizes[4];       // 2
  const int S    = ntok / B;          // 2048

  float* pc = (float*)d_ws;
  float* pw = pc + ntok;
  float* pv = pw + ntok;
  float* pb = pv + ntok;
  float* pr = pb + ntok;

  rcl_row_kernel<<<ntok, TPB, 0, stream>>>(logits, labels, bond, rup, ps,
                                           pc, pw, pv, pb, pr, V, S);
  rcl_final_kernel<<<1, TPB, 0, stream>>>(pc, pw, pv, pb, pr, out,
                                          ntok, 1.0f / (float)B);
}